// GraphAttentionBlock_18511309045799
// MI455X (gfx1250) — compile-verified
//
#include <hip/hip_runtime.h>
#include <math.h>

// ---------------------------------------------------------------------------
// Types for CDNA5 WMMA (wave32).
// ---------------------------------------------------------------------------
typedef __attribute__((ext_vector_type(2))) float v2f;
typedef __attribute__((ext_vector_type(8))) float v8f;

__device__ inline v8f wmma_f32_16x16x4(v2f a, v2f b, v8f c) {
  return __builtin_amdgcn_wmma_f32_16x16x4_f32(false, a, false, b, (short)0, c,
                                               false, false);
}

__device__ inline v8f v8f_zero() {
  v8f z = {0.f, 0.f, 0.f, 0.f, 0.f, 0.f, 0.f, 0.f};
  return z;
}

// ---------------------------------------------------------------------------
// CDNA5 async global->LDS staging (ASYNCcnt), with synchronous fallback.
// Builtin param types per hipcc diagnostics:
//   b32 : (AS1 int*, AS3 int*, imm int, imm int)
//   b128: (AS1 vector_size(16) int*, AS3 vector_size(16) int*, imm, imm)
// ---------------------------------------------------------------------------
#define AS1 __attribute__((address_space(1)))
#define AS3 __attribute__((address_space(3)))

typedef int gvi4 __attribute__((vector_size(16)));  // GCC-style vector (match)

#if defined(__has_builtin)
#if __has_builtin(__builtin_amdgcn_global_load_async_to_lds_b32) &&     \
    __has_builtin(__builtin_amdgcn_global_load_async_to_lds_b128) &&    \
    __has_builtin(__builtin_amdgcn_s_wait_asynccnt)
#define HAVE_ASYNC_LDS 1
#endif
#endif

__device__ inline void copy4(const float* g, float* l) {
#ifdef HAVE_ASYNC_LDS
  __builtin_amdgcn_global_load_async_to_lds_b32((AS1 int*)(g), (AS3 int*)(l),
                                                0, 0);
#else
  *l = *g;
#endif
}

__device__ inline void copy16(const float* g, float* l) {
#ifdef HAVE_ASYNC_LDS
  __builtin_amdgcn_global_load_async_to_lds_b128((AS1 gvi4*)(g),
                                                 (AS3 gvi4*)(l), 0, 0);
#else
  *(float4*)l = *(const float4*)g;
#endif
}

__device__ inline void wait_async0() {
#ifdef HAVE_ASYNC_LDS
  __builtin_amdgcn_s_wait_asynccnt(0);
#endif
}

#define HIDDEN 256
#define BATCH 8
#define BN_EPS 1e-5f

// ---------------------------------------------------------------------------
// Kernel 0: zero the node accumulator (we pool via atomics).
// ---------------------------------------------------------------------------
__global__ void zero_nodes_kernel(float* __restrict__ nodes, int n) {
  int i = blockIdx.x * blockDim.x + threadIdx.x;
  if (i < n) nodes[i] = 0.f;
}

// ---------------------------------------------------------------------------
// Kernel 1: fused 3x3 conv (C->256, pad 1) + bias + BN + exact GELU + global
// average pool. Implicit GEMM, K = C*9 in 36-wide chunks (4 channels).
// Workgroup = 256 threads = 8 waves, owns 128 output channels (half) x 16
// pixels; each wave one 16x16 tile. Weights + im2col patch double-buffered in
// LDS (weights via async-load-to-LDS when available), one barrier per chunk.
// ---------------------------------------------------------------------------
template <int C, int S>
__global__ __launch_bounds__(256) void conv3x3_gelu_pool_kernel(
    const float* __restrict__ x,      // [B, C, S, S]
    const float* __restrict__ w,      // [256, C, 3, 3]
    const float* __restrict__ bias, const float* __restrict__ gbn,
    const float* __restrict__ beta, const float* __restrict__ mean,
    const float* __restrict__ var,
    float* __restrict__ nodes_raw,    // [B, 4, 256]
    int scale) {
  constexpr int TPR = S / 16;  // 16-pixel tiles per row
  constexpr int KC = 36;       // K per chunk (4 channels x 9 taps)
  constexpr int MH = 128;      // output channels per workgroup
  constexpr int K9 = C * 9;

  int idx = blockIdx.x;
  const int half = idx & 1;
  idx >>= 1;
  const int b = idx / (S * TPR);
  int rem = idx - b * (S * TPR);
  const int oh = rem / TPR;
  const int ow0 = (rem - oh * TPR) * 16;

  __shared__ float wbuf[2][MH * KC];  // [m][k], bank-conflict-free stride 36
  __shared__ float pbuf[2][KC * 16];  // [k][n] im2col patch
  __shared__ float red[MH];

  const int t = threadIdx.x;
  const int lane = t & 31;
  const int wave = t >> 5;
  const int nlane = lane & 15;
  const int hi = lane >> 4;  // f32 WMMA: lanes>=16 supply K+2/K+3
  const int dlt = hi * 2;

  const float* wbase = w + (size_t)(half * MH) * K9;

  v8f acc = v8f_zero();

  auto stage = [&](int buf, int c0) {
    // weight chunk: MH rows x 36 k, float4 granularity (coalesced / async)
    for (int j = t; j < MH * 9; j += 256) {
      int m = j / 9;
      int k4 = j - m * 9;
      copy16(wbase + (size_t)m * K9 + c0 * 9 + k4 * 4,
             &wbuf[buf][m * KC + k4 * 4]);
    }
    // im2col patch with zero padding (needs bounds -> synchronous)
    for (int j = t; j < KC * 16; j += 256) {
      int k = j >> 4;
      int n = j & 15;
      int ci = k / 9;
      int r = k - ci * 9;
      int kh = r / 3;
      int kw = r - kh * 3;
      int ih = oh + kh - 1;
      int iw = ow0 + n + kw - 1;
      float v = 0.f;
      if ((unsigned)ih < (unsigned)S && (unsigned)iw < (unsigned)S)
        v = x[(((size_t)b * C + (c0 + ci)) * S + ih) * S + iw];
      pbuf[buf][j] = v;
    }
  };

  auto compute = [&](int buf) {
    const int mrow = wave * 16 + nlane;
    const float* wl = &wbuf[buf][mrow * KC];
    const float* pl = pbuf[buf];
#pragma unroll
    for (int kk = 0; kk < KC; kk += 4) {
      int k0 = kk + dlt;
      v2f a;
      a.x = wl[k0];
      a.y = wl[k0 + 1];
      v2f bb;
      bb.x = pl[k0 * 16 + nlane];
      bb.y = pl[(k0 + 1) * 16 + nlane];
      acc = wmma_f32_16x16x4(a, bb, acc);
    }
  };

  stage(0, 0);
  wait_async0();
  __syncthreads();
  int cur = 0;
  for (int c0 = 0; c0 < C; c0 += 4) {
    if (c0 + 4 < C) stage(cur ^ 1, c0 + 4);  // pipeline next chunk
    compute(cur);
    wait_async0();
    __syncthreads();
    cur ^= 1;
  }

  // fused epilogue: bias + BN + exact GELU + pool (LDS reduce, then atomics)
  if (t < MH) red[t] = 0.f;
  __syncthreads();
  const int Mb = wave * 16;
#pragma unroll
  for (int v = 0; v < 8; ++v) {
    int m = Mb + v + 8 * hi;  // C/D layout: vgpr v -> row v (+8 for hi lanes)
    int gm = half * MH + m;
    float pre = acc[v] + bias[gm];
    float invs = gbn[gm] * rsqrtf(var[gm] + BN_EPS);
    float bn = (pre - mean[gm]) * invs + beta[gm];
    float ge = 0.5f * bn * (1.0f + erff(bn * 0.70710678118654752440f));
    atomicAdd(&red[m], ge);  // ds_add_f32
  }
  __syncthreads();
  if (t < MH) {
    atomicAdd(&nodes_raw[((size_t)b * 4 + scale) * 256 + half * MH + t],
              red[t] * (1.0f / (float)(S * S)));
  }
}

// ---------------------------------------------------------------------------
// Kernel 2: both GAT layers (tiny: B=8, N=4, H=256). One block per batch.
// ---------------------------------------------------------------------------
__global__ __launch_bounds__(256) void gat_kernel(
    const float* __restrict__ nodes_raw, const float* __restrict__ pos_enc,
    const float* __restrict__ W1, const float* __restrict__ as1,
    const float* __restrict__ ad1, const float* __restrict__ b1,
    const float* __restrict__ W2, const float* __restrict__ as2,
    const float* __restrict__ ad2, const float* __restrict__ b2,
    float* __restrict__ nodes_out) {
  int b = blockIdx.x;
  int t = threadIdx.x;
  __shared__ float xn[4][256], xh[4][256], x1s[4][256], xh2[4][256];
  __shared__ float s1[4][8], d1v[4][8], alpha1[4][4][8];
  __shared__ float s2[4], d2v[4], alpha2[4][4];

  for (int n = 0; n < 4; ++n)
    xn[n][t] = nodes_raw[((size_t)b * 4 + n) * 256 + t] + pos_enc[n * 256 + t];
  __syncthreads();

  for (int n = 0; n < 4; ++n) {
    float s = 0.f;
    for (int h = 0; h < 256; ++h) s += xn[n][h] * W1[h * 256 + t];
    xh[n][t] = s;
  }
  __syncthreads();

  if (t < 32) {
    int n = t >> 3, k = t & 7;
    float ss = 0.f, dd = 0.f;
    for (int d = 0; d < 32; ++d) {
      float v = xh[n][k * 32 + d];
      ss += v * as1[k * 32 + d];
      dd += v * ad1[k * 32 + d];
    }
    s1[n][k] = ss;
    d1v[n][k] = dd;
  }
  __syncthreads();

  if (t < 32) {
    int i = t >> 3, k = t & 7;
    float l[4], mx = -1e30f;
    for (int j = 0; j < 4; ++j) {
      float v = s1[j][k] + d1v[i][k];
      v = (v > 0.f) ? v : 0.2f * v;  // leaky_relu(0.2)
      l[j] = v;
      mx = fmaxf(mx, v);
    }
    float sum = 0.f;
    for (int j = 0; j < 4; ++j) {
      l[j] = expf(l[j] - mx);
      sum += l[j];
    }
    float rs = 1.f / sum;
    for (int j = 0; j < 4; ++j) alpha1[i][j][k] = l[j] * rs;
  }
  __syncthreads();

  int k8 = t >> 5;
  for (int i = 0; i < 4; ++i) {
    float s = b1[t];
    for (int j = 0; j < 4; ++j) s += alpha1[i][j][k8] * xh[j][t];
    x1s[i][t] = fmaxf(s, 0.f);
  }
  __syncthreads();

  for (int n = 0; n < 4; ++n) {
    float s = 0.f;
    for (int h = 0; h < 256; ++h) s += x1s[n][h] * W2[h * 256 + t];
    xh2[n][t] = s;
  }
  __syncthreads();
  if (t < 8) {
    int n = t & 3;
    const float* a = (t < 4) ? as2 : ad2;
    float acc = 0.f;
    for (int d = 0; d < 256; ++d) acc += xh2[n][d] * a[d];
    if (t < 4)
      s2[n] = acc;
    else
      d2v[n] = acc;
  }
  __syncthreads();
  if (t < 4) {
    int i = t;
    float l[4], mx = -1e30f;
    for (int j = 0; j < 4; ++j) {
      float v = s2[j] + d2v[i];
      v = (v > 0.f) ? v : 0.2f * v;
      l[j] = v;
      mx = fmaxf(mx, v);
    }
    float sum = 0.f;
    for (int j = 0; j < 4; ++j) {
      l[j] = expf(l[j] - mx);
      sum += l[j];
    }
    float rs = 1.f / sum;
    for (int j = 0; j < 4; ++j) alpha2[i][j] = l[j] * rs;
  }
  __syncthreads();
  for (int i = 0; i < 4; ++i) {
    float s = b2[t];
    for (int j = 0; j < 4; ++j) s += alpha2[i][j] * xh2[j][t];
    nodes_out[((size_t)b * 4 + i) * 256 + t] = fmaxf(s, 0.f);
  }
}

// ---------------------------------------------------------------------------
// Kernel 3: node half of the 1x1 conv: z[b][o] = fbias[o] + W[:, :256] @ node.
// ---------------------------------------------------------------------------
__global__ __launch_bounds__(512) void node_proj_kernel(
    const float* __restrict__ x2,  // [B,4,256]
    const float* __restrict__ w0, const float* __restrict__ fb0,
    const float* __restrict__ w1, const float* __restrict__ fb1,
    const float* __restrict__ w2, const float* __restrict__ fb2,
    const float* __restrict__ w3, const float* __restrict__ fb3,
    float* __restrict__ z) {
  int b = blockIdx.x, sc = blockIdx.y, t = threadIdx.x;
  int C;
  const float* w;
  const float* fb;
  int zoff;
  switch (sc) {
    case 0: C = 64;  w = w0; fb = fb0; zoff = 0;                        break;
    case 1: C = 128; w = w1; fb = fb1; zoff = BATCH * 64;               break;
    case 2: C = 256; w = w2; fb = fb2; zoff = BATCH * (64 + 128);       break;
    default:C = 512; w = w3; fb = fb3; zoff = BATCH * (64 + 128 + 256); break;
  }
  if (t < C) {
    const float* xr = x2 + ((size_t)b * 4 + sc) * 256;
    int stride = 256 + C;
    float s = fb[t];
    for (int h = 0; h < 256; ++h) s += w[(size_t)t * stride + h] * xr[h];
    z[zoff + b * C + t] = s;
  }
}

// ---------------------------------------------------------------------------
// Kernel 4: fused 1x1 conv (skip half) + BN + residual.
// GEMM M=C, N=16 pixels/block, K=C; skip tile AND weight chunk staged in LDS
// (async when available). wlds rows padded to 20 floats (bank-conflict-free).
// ---------------------------------------------------------------------------
template <int C, int S>
__global__ __launch_bounds__(256) void fuse1x1_kernel(
    const float* __restrict__ skip,  // [B, C, S*S]
    const float* __restrict__ wf,    // [C, 256+C]
    const float* __restrict__ gbn, const float* __restrict__ beta,
    const float* __restrict__ mean, const float* __restrict__ var,
    const float* __restrict__ z,     // [B, C] (includes conv bias)
    float* __restrict__ out) {
  constexpr int SS = S * S;
  constexpr int PT = SS / 16;
  constexpr int MS = C / 16;        // M subtiles
  constexpr int NS = (MS + 7) / 8;  // subtiles per wave
  constexpr int WP = 20;            // padded LDS row stride
  int idx = blockIdx.x;
  const int b = idx / PT;
  const int p0 = (idx - b * PT) * 16;

  __shared__ float bt[16 * 16];
  __shared__ float wlds[C * WP];

  const int t = threadIdx.x;
  const int lane = t & 31;
  const int wave = t >> 5;
  const int nlane = lane & 15;
  const int hi = lane >> 4;
  const int dlt = hi * 2;
  const int stride = 256 + C;

  v8f acc[NS];
#pragma unroll
  for (int s = 0; s < NS; ++s) acc[s] = v8f_zero();

  for (int c0 = 0; c0 < C; c0 += 16) {
    {  // stage skip tile (K x N = 16 x 16), one b32 per thread
      int k = t >> 4, n = t & 15;
      copy4(&skip[((size_t)b * C + (c0 + k)) * SS + p0 + n], &bt[t]);
    }
    // stage weight chunk: C rows x 16 k as float4s
    for (int j = t; j < C * 4; j += 256) {
      int m = j >> 2;
      int k4 = j & 3;
      copy16(&wf[(size_t)m * stride + 256 + c0 + k4 * 4],
             &wlds[m * WP + k4 * 4]);
    }
    wait_async0();
    __syncthreads();
#pragma unroll
    for (int s = 0; s < NS; ++s) {
      int msub = wave + s * 8;
      if (msub < MS) {  // wave-uniform
        int m = msub * 16 + nlane;
        const float* wl = &wlds[m * WP];
#pragma unroll
        for (int kk = 0; kk < 16; kk += 4) {
          int k0 = kk + dlt;
          v2f a;
          a.x = wl[k0];
          a.y = wl[k0 + 1];
          v2f bb;
          bb.x = bt[k0 * 16 + nlane];
          bb.y = bt[(k0 + 1) * 16 + nlane];
          acc[s] = wmma_f32_16x16x4(a, bb, acc[s]);
        }
      }
    }
    __syncthreads();
  }

#pragma unroll
  for (int s = 0; s < NS; ++s) {
    int msub = wave + s * 8;
    if (msub < MS) {
      int Mb = msub * 16;
#pragma unroll
      for (int v = 0; v < 8; ++v) {
        int m = Mb + v + 8 * hi;
        int p = p0 + nlane;
        float pre = acc[s][v] + z[b * C + m];
        float invs = gbn[m] * rsqrtf(var[m] + BN_EPS);
        float val = (pre - mean[m]) * invs + beta[m];
        size_t off = ((size_t)b * C + m) * SS + p;
        out[off] = skip[off] + val;
      }
    }
  }
}

// ---------------------------------------------------------------------------
// Host: pointer mapping + launches.
// ---------------------------------------------------------------------------
extern "C" void kernel_launch(void* const* d_in, const int* in_sizes, int n_in,
                              void* d_out, int out_size, void* d_ws,
                              size_t ws_size, hipStream_t stream) {
  auto f = [&](int i) { return (const float*)d_in[i]; };

  const float *skip[4], *pos_enc;
  const float *pw[4], *pb[4], *pg[4], *pbe[4], *pme[4], *pva[4];
  const float *g1W, *g1as, *g1ad, *g1b, *g2W, *g2as, *g2ad, *g2b;
  const float *fw[4], *fb[4], *fg[4], *fbe[4], *fme[4], *fva[4];

  if (in_sizes[0] == 8 * 64 * 128 * 128) {
    // Insertion-order flattening: skips, pos_enc, proj[], gat1, gat2, final[]
    for (int i = 0; i < 4; ++i) skip[i] = f(i);
    pos_enc = f(4);
    for (int i = 0; i < 4; ++i) {
      pw[i] = f(5 + 6 * i);  pb[i] = f(6 + 6 * i);  pg[i] = f(7 + 6 * i);
      pbe[i] = f(8 + 6 * i); pme[i] = f(9 + 6 * i); pva[i] = f(10 + 6 * i);
    }
    g1W = f(29); g1as = f(30); g1ad = f(31); g1b = f(32);
    g2W = f(33); g2as = f(34); g2ad = f(35); g2b = f(36);
    for (int i = 0; i < 4; ++i) {
      fw[i] = f(37 + 6 * i);  fb[i] = f(38 + 6 * i); fg[i] = f(39 + 6 * i);
      fbe[i] = f(40 + 6 * i); fme[i] = f(41 + 6 * i); fva[i] = f(42 + 6 * i);
    }
  } else {
    // jax tree_flatten (sorted keys): final[], gat1, gat2, pos_enc, proj[], skips
    for (int i = 0; i < 4; ++i) {
      fb[i] = f(0 + 6 * i);  fbe[i] = f(1 + 6 * i); fg[i] = f(2 + 6 * i);
      fme[i] = f(3 + 6 * i); fva[i] = f(4 + 6 * i); fw[i] = f(5 + 6 * i);
    }
    g1W = f(24); g1ad = f(25); g1as = f(26); g1b = f(27);
    g2W = f(28); g2ad = f(29); g2as = f(30); g2b = f(31);
    pos_enc = f(32);
    for (int i = 0; i < 4; ++i) {
      pb[i] = f(33 + 6 * i);  pbe[i] = f(34 + 6 * i); pg[i] = f(35 + 6 * i);
      pme[i] = f(36 + 6 * i); pva[i] = f(37 + 6 * i); pw[i] = f(38 + 6 * i);
    }
    for (int i = 0; i < 4; ++i) skip[i] = f(57 + i);
  }

  float* ws = (float*)d_ws;
  float* nodes_raw = ws;         // [B,4,256]
  float* nodes_out = ws + 8192;  // [B,4,256]
  float* zbuf = ws + 16384;      // packed z
  float* out = (float*)d_out;

  zero_nodes_kernel<<<32, 256, 0, stream>>>(nodes_raw, BATCH * 4 * 256);

  // fused conv3x3 + BN + GELU + pool (2 workgroups per pixel tile: M halves)
  conv3x3_gelu_pool_kernel<64, 128><<<BATCH * 128 * 8 * 2, 256, 0, stream>>>(
      skip[0], pw[0], pb[0], pg[0], pbe[0], pme[0], pva[0], nodes_raw, 0);
  conv3x3_gelu_pool_kernel<128, 64><<<BATCH * 64 * 4 * 2, 256, 0, stream>>>(
      skip[1], pw[1], pb[1], pg[1], pbe[1], pme[1], pva[1], nodes_raw, 1);
  conv3x3_gelu_pool_kernel<256, 32><<<BATCH * 32 * 2 * 2, 256, 0, stream>>>(
      skip[2], pw[2], pb[2], pg[2], pbe[2], pme[2], pva[2], nodes_raw, 2);
  conv3x3_gelu_pool_kernel<512, 16><<<BATCH * 16 * 1 * 2, 256, 0, stream>>>(
      skip[3], pw[3], pb[3], pg[3], pbe[3], pme[3], pva[3], nodes_raw, 3);

  gat_kernel<<<BATCH, 256, 0, stream>>>(nodes_raw, pos_enc, g1W, g1as, g1ad,
                                        g1b, g2W, g2as, g2ad, g2b, nodes_out);

  node_proj_kernel<<<dim3(BATCH, 4), 512, 0, stream>>>(
      nodes_out, fw[0], fb[0], fw[1], fb[1], fw[2], fb[2], fw[3], fb[3], zbuf);

  const int zo0 = 0, zo1 = BATCH * 64, zo2 = BATCH * (64 + 128),
            zo3 = BATCH * (64 + 128 + 256);
  const size_t oo0 = 0;
  const size_t oo1 = oo0 + (size_t)BATCH * 64 * 128 * 128;
  const size_t oo2 = oo1 + (size_t)BATCH * 128 * 64 * 64;
  const size_t oo3 = oo2 + (size_t)BATCH * 256 * 32 * 32;

  fuse1x1_kernel<64, 128><<<BATCH * (128 * 128 / 16), 256, 0, stream>>>(
      skip[0], fw[0], fg[0], fbe[0], fme[0], fva[0], zbuf + zo0, out + oo0);
  fuse1x1_kernel<128, 64><<<BATCH * (64 * 64 / 16), 256, 0, stream>>>(
      skip[1], fw[1], fg[1], fbe[1], fme[1], fva[1], zbuf + zo1, out + oo1);
  fuse1x1_kernel<256, 32><<<BATCH * (32 * 32 / 16), 256, 0, stream>>>(
      skip[2], fw[2], fg[2], fbe[2], fme[2], fva[2], zbuf + zo2, out + oo2);
  fuse1x1_kernel<512, 16><<<BATCH * (16 * 16 / 16), 256, 0, stream>>>(
      skip[3], fw[3], fg[3], fbe[3], fme[3], fva[3], zbuf + zo3, out + oo3);
}